// Img2WordsCNN_55637006352791
// MI455X (gfx1250) — compile-verified
//
#include <hip/hip_runtime.h>
#include <stdint.h>

// ---------------------------------------------------------------------------
// Img2WordsCNN top-k word emitter for MI455X (gfx1250).
//   in : mul_class [B, VOCAB] fp32,  seqlen [1] int32 (=50)
//   out: [B*VOCAB] fp32 copy of mul_class, then [B*L] int32 words
// One workgroup (512 threads / 16 wave32) per row.
// All bulk reads go through the Tensor Data Mover (tensor_load_to_lds +
// s_wait_tensorcnt, double buffered):
//   pass 1: copy row to output + 1024-bin LDS histogram (ds_add_u32)
//   pass 2: gather >=threshold candidates from staged LDS tiles
// then LDS bitonic sort (descending value, ascending index = top_k stable)
// and a prefix-valid mask to emit the word table.
// ---------------------------------------------------------------------------

#define VOCAB     50257
#define BS        512
#define NBINS     1024
#define CAP       4096      // candidate buffer (power of two)
#define TILE      4096      // elements per TDM tile (16 KB)
#define PAD_WORD  2

typedef unsigned int u32;
typedef u32 v4u __attribute__((ext_vector_type(4)));
typedef int  v4i __attribute__((ext_vector_type(4)));
typedef int  v8i __attribute__((ext_vector_type(8)));

// Issue a 1-D TDM load: `tile_elems` fp32 from gptr to the LDS byte address in lds_byte_off.
// tensor_dim0 = remaining elements, so the out-of-tensor tail reads zero.
__device__ __forceinline__ void tdm_load_row_tile(u32 lds_byte_off,
                                                  const float* gptr,
                                                  u32 remaining_elems,
                                                  u32 tile_elems) {
  unsigned long long ga = (unsigned long long)(size_t)gptr;
  v4u g0;
  g0.x = 1u;                                                // count=1, user desc
  g0.y = lds_byte_off;                                      // lds_addr (bytes)
  g0.z = (u32)(ga & 0xFFFFFFFFull);                         // global_addr[31:0]
  g0.w = (u32)((ga >> 32) & 0x01FFFFFFull) | (2u << 30);    // [56:32] + type=2
  v8i g1;
  g1[0] = (int)(2u << 16);                                  // data_size = 4B
  g1[1] = (int)((remaining_elems & 0xFFFFu) << 16);         // tensor_dim0[15:0]
  g1[2] = (int)((remaining_elems >> 16) & 0xFFFFu);         // tensor_dim0[31:16]
  g1[3] = (int)((tile_elems & 0xFFFFu) << 16);              // tile_dim0
  g1[4] = 0; g1[5] = 0; g1[6] = 0; g1[7] = 0;               // strides unused (1-D)
  v4i z4 = {0, 0, 0, 0};
#if __clang_major__ >= 23
  v8i z8 = {0, 0, 0, 0, 0, 0, 0, 0};
  __builtin_amdgcn_tensor_load_to_lds(g0, g1, z4, z4, z8, 0);
#else
  __builtin_amdgcn_tensor_load_to_lds(g0, g1, z4, z4, 0);
#endif
}

__global__ __launch_bounds__(BS)
void img2words_topk_kernel(const float* __restrict__ in,
                           const int*  __restrict__ seqlen,
                           float* __restrict__ outv,
                           int*   __restrict__ outw) {
  __shared__ float stage[2][TILE];
  __shared__ int   hist[NBINS];
  __shared__ float cv[CAP];
  __shared__ int   ci[CAP];
  __shared__ int   s_bthresh;
  __shared__ int   s_candn;
  __shared__ int   s_n;
  __shared__ int   s_P;

  const int    tid    = threadIdx.x;
  const int    row    = blockIdx.x;
  const size_t rowoff = (size_t)row * VOCAB;
  const float* rowp   = in + rowoff;
  const int    K      = min(seqlen[0], VOCAB);
  const int    NT     = (VOCAB + TILE - 1) / TILE;

  for (int i = tid; i < NBINS; i += BS) hist[i] = 0;
  if (tid == 0) s_candn = 0;
  __syncthreads();

  // ---- Pass 1: TDM-staged stream: copy row to outv + LDS histogram --------
  if (tid < 32) {  // wave 0 drives the DMA pipeline
    tdm_load_row_tile((u32)(size_t)&stage[0][0], rowp, (u32)VOCAB, TILE);
  }
  for (int t = 0; t < NT; ++t) {
    if (tid < 32) {
      if (t + 1 < NT) {
        tdm_load_row_tile((u32)(size_t)&stage[(t + 1) & 1][0],
                          rowp + (size_t)(t + 1) * TILE,
                          (u32)(VOCAB - (t + 1) * TILE), TILE);
        __builtin_amdgcn_s_wait_tensorcnt((short)1);   // tile t landed
      } else {
        __builtin_amdgcn_s_wait_tensorcnt((short)0);
      }
    }
    __syncthreads();                                    // tile t visible to WG
    const int    base = t * TILE;
    const float* buf  = stage[t & 1];
    for (int i = tid; i < TILE; i += BS) {
      int g = base + i;
      if (g < VOCAB) {
        float v = buf[i];
        outv[rowoff + g] = v;
        int bin = (int)(v * (float)NBINS);
        bin = bin < 0 ? 0 : (bin > NBINS - 1 ? NBINS - 1 : bin);
        atomicAdd(&hist[bin], 1);                       // ds_add_u32
      }
    }
    __syncthreads();                                    // buffer reusable
  }

  // ---- Threshold bin = top of histogram holding >= K values ---------------
  // (uniform data: ~49 values/bin, loop exits after ~2 iterations)
  if (tid == 0) {
    int acc = 0, b = 0;
    for (int bin = NBINS - 1; bin >= 0; --bin) {
      acc += hist[bin];
      if (acc >= K) { b = bin; break; }
    }
    s_bthresh = b;
  }
  __syncthreads();
  const int bthr = s_bthresh;

  // ---- Pass 2: TDM-staged candidate gather (tiles now L2-resident) --------
  if (tid < 32) {
    tdm_load_row_tile((u32)(size_t)&stage[0][0], rowp, (u32)VOCAB, TILE);
  }
  for (int t = 0; t < NT; ++t) {
    if (tid < 32) {
      if (t + 1 < NT) {
        tdm_load_row_tile((u32)(size_t)&stage[(t + 1) & 1][0],
                          rowp + (size_t)(t + 1) * TILE,
                          (u32)(VOCAB - (t + 1) * TILE), TILE);
        __builtin_amdgcn_s_wait_tensorcnt((short)1);
      } else {
        __builtin_amdgcn_s_wait_tensorcnt((short)0);
      }
    }
    __syncthreads();
    const int    base = t * TILE;
    const float* buf  = stage[t & 1];
    for (int i = tid; i < TILE; i += BS) {
      int g = base + i;
      if (g < VOCAB) {
        float v = buf[i];
        int bin = (int)(v * (float)NBINS);
        bin = bin < 0 ? 0 : (bin > NBINS - 1 ? NBINS - 1 : bin);
        if (bin >= bthr) {
          int pos = atomicAdd(&s_candn, 1);
          if (pos < CAP) { cv[pos] = v; ci[pos] = g; }
        }
      }
    }
    __syncthreads();
  }

  // ---- Bitonic sort, descending value / ascending index -------------------
  if (tid == 0) {
    int n = s_candn < CAP ? s_candn : CAP;
    int P = 64;
    while (P < n) P <<= 1;
    s_n = n; s_P = P;
  }
  __syncthreads();
  const int n = s_n, P = s_P;
  for (int i = n + tid; i < P; i += BS) { cv[i] = -__builtin_inff(); ci[i] = 0x7FFFFFFF; }
  __syncthreads();
  for (int k = 2; k <= P; k <<= 1) {
    for (int j = k >> 1; j > 0; j >>= 1) {
      for (int i = tid; i < P; i += BS) {
        int ixj = i ^ j;
        if (ixj > i) {
          float v1 = cv[i], v2 = cv[ixj];
          int   i1 = ci[i], i2 = ci[ixj];
          bool dir   = ((i & k) == 0);
          // "i ranks after ixj" in the final descending / stable order
          bool after = (v1 < v2) || (v1 == v2 && i1 > i2);
          if (after == dir) { cv[i] = v2; cv[ixj] = v1; ci[i] = i2; ci[ixj] = i1; }
        }
      }
      __syncthreads();
    }
  }

  // ---- Prefix-valid mask, emit words --------------------------------------
  if (tid == 0) {
    int valid = 1;
    for (int i = 0; i < K; ++i) {
      float v   = (i < n) ? cv[i] : 0.0f;
      int   idx = (i < n) ? ci[i] : PAD_WORD;
      valid &= ((int)(v * 2.0f) != 0) ? 1 : 0;   // nonzero iff v >= 0.5
      outw[(size_t)row * K + i] = valid ? idx : PAD_WORD;
    }
  }
}

extern "C" void kernel_launch(void* const* d_in, const int* in_sizes, int n_in,
                              void* d_out, int out_size, void* d_ws, size_t ws_size,
                              hipStream_t stream) {
  (void)n_in; (void)out_size; (void)d_ws; (void)ws_size;
  const float* in     = (const float*)d_in[0];
  const int*   seqlen = (const int*)d_in[1];
  const int    B      = in_sizes[0] / VOCAB;

  float* outv = (float*)d_out;                       // [B*VOCAB] fp32 copy
  int*   outw = (int*)d_out + (size_t)B * VOCAB;     // [B*L] int32 words

  img2words_topk_kernel<<<dim3(B), dim3(BS), 0, stream>>>(in, seqlen, outv, outw);
}